// CoAtt_22539988369716
// MI455X (gfx1250) — compile-verified
//
#include <hip/hip_runtime.h>
#include <math.h>

typedef __attribute__((ext_vector_type(2))) float v2f;
typedef __attribute__((ext_vector_type(4))) float v4f;
typedef __attribute__((ext_vector_type(8))) float v8f;

namespace {
constexpr int kB = 2048;
constexpr int kS = 64;
constexpr int kH = 256;
constexpr int kD = 256;
constexpr int XS_STRIDE = 260;            // 256 + 4 pad: conflict-free WMMA fragment gathers
constexpr float kNullAtt = -4194304.0f;   // -2^22

// dynamic LDS layout (float offsets)
constexpr int OFF_XS    = 0;                       // [64][260] staged x_session
constexpr int OFF_Q     = OFF_XS + kS * XS_STRIDE; // [64][260] query
constexpr int OFF_ITEM  = OFF_Q + kS * XS_STRIDE;  // [256]
constexpr int OFF_IPROJ = OFF_ITEM + kD;           // [256] item@W1[:, :256] + b1
constexpr int OFF_SCORE = OFF_IPROJ + kD;          // [256]
constexpr int OFF_W     = OFF_SCORE + kH;          // [256] softmax weights
constexpr int OFF_REDA  = OFF_W + kH;              // [8]
constexpr int OFF_REDB  = OFF_REDA + 8;            // [8]
constexpr int LDS_FLOATS = OFF_REDB + 8;
}

__global__ __launch_bounds__(256)
void coatt_wmma_kernel(const float* __restrict__ item_emb,
                       const float* __restrict__ x_session,
                       const int*   __restrict__ session_len,
                       const float* __restrict__ user_hist,
                       const int*   __restrict__ hist_len,
                       const float* __restrict__ W1,
                       const float* __restrict__ b1,
                       float* __restrict__ out)   // [B*D rep | B*H score]
{
  extern __shared__ float lds[];
  float* sm_xs    = lds + OFF_XS;
  float* sm_q     = lds + OFF_Q;
  float* sm_item  = lds + OFF_ITEM;
  float* sm_iproj = lds + OFF_IPROJ;
  float* sm_score = lds + OFF_SCORE;
  float* sm_w     = lds + OFF_W;
  float* sm_redA  = lds + OFF_REDA;
  float* sm_redB  = lds + OFF_REDB;

  const int b  = blockIdx.x;
  const int t  = threadIdx.x;   // 0..255
  const int w  = t >> 5;        // wave id 0..7
  const int ln = t & 31;        // lane in wave32
  const int hl = ln & 15;       // lane within half-wave
  const int hi = ln >> 4;       // 0: lanes 0-15, 1: lanes 16-31

  const float* xs   = x_session + (size_t)b * kS * kD;
  const float* hist = user_hist + (size_t)b * kH * kD;

  // ---- Phase 0: stage x_session (coalesced float4) and item_emb into LDS ----
  sm_item[t] = item_emb[(size_t)b * kD + t];
  for (int e = t * 4; e < kS * kD; e += 256 * 4) {
    v4f v = *(const v4f*)(xs + e);
    int row = e >> 8;
    int col = e & 255;
    *(v4f*)(sm_xs + row * XS_STRIDE + col) = v;
  }
  __syncthreads();

  // ---- Phase 1: iproj[d] = item . W1[d, 0:256] + b1[d]  (s-invariant half) ----
  {
    float acc = 0.f;
    const float* wrow = W1 + (size_t)t * (2 * kD);
    #pragma unroll 4
    for (int k = 0; k < kD; k += 4) {
      v4f wv = *(const v4f*)(wrow + k);
      acc += sm_item[k] * wv.x + sm_item[k + 1] * wv.y
           + sm_item[k + 2] * wv.z + sm_item[k + 3] * wv.w;
    }
    sm_iproj[t] = acc + b1[t];
  }
  __syncthreads();

  // ---- Phase 2: query[s,d] = iproj[d] + sum_k xs[s,k] * W1[d, 256+k]  (WMMA f32) ----
  // Each wave owns 2 d-tiles (N) x 4 s-tiles (M); K loop in steps of 4.
  for (int nt = 2 * w; nt < 2 * w + 2; ++nt) {
    v8f acc0 = {}, acc1 = {}, acc2 = {}, acc3 = {};
    // B fragment: B[k][n] = W1[(nt*16+n)*512 + 256 + k]; lane: n=hl, K pair = k0 + 2*hi
    const float* wbase = W1 + (size_t)(nt * 16 + hl) * (2 * kD) + kD + (hi << 1);
    // A fragment: A[m][k] = sm_xs[row][k]; lane: row = m*16+hl, K pair = k0 + 2*hi
    const float* abase = sm_xs + hl * XS_STRIDE + (hi << 1);
    for (int k0 = 0; k0 < kD; k0 += 4) {
      v2f bf = *(const v2f*)(wbase + k0);
      v2f a0 = *(const v2f*)(abase + 0 * 16 * XS_STRIDE + k0);
      v2f a1 = *(const v2f*)(abase + 1 * 16 * XS_STRIDE + k0);
      v2f a2 = *(const v2f*)(abase + 2 * 16 * XS_STRIDE + k0);
      v2f a3 = *(const v2f*)(abase + 3 * 16 * XS_STRIDE + k0);
      acc0 = __builtin_amdgcn_wmma_f32_16x16x4_f32(false, a0, false, bf, (short)0, acc0, false, false);
      acc1 = __builtin_amdgcn_wmma_f32_16x16x4_f32(false, a1, false, bf, (short)0, acc1, false, false);
      acc2 = __builtin_amdgcn_wmma_f32_16x16x4_f32(false, a2, false, bf, (short)0, acc2, false, false);
      acc3 = __builtin_amdgcn_wmma_f32_16x16x4_f32(false, a3, false, bf, (short)0, acc3, false, false);
    }
    // C/D layout: VGPR r, lane -> (s = m*16 + r + 8*hi, d = nt*16 + hl)
    float ip = sm_iproj[nt * 16 + hl];
    int col = nt * 16 + hl;
    #pragma unroll
    for (int r = 0; r < 8; ++r) {
      int srow = (hi << 3) + r;
      sm_q[(srow + 0 ) * XS_STRIDE + col] = acc0[r] + ip;
      sm_q[(srow + 16) * XS_STRIDE + col] = acc1[r] + ip;
      sm_q[(srow + 32) * XS_STRIDE + col] = acc2[r] + ip;
      sm_q[(srow + 48) * XS_STRIDE + col] = acc3[r] + ip;
    }
  }
  __syncthreads();

  // ---- Phase 3: att[s,h] = query[s,:] . hist[h,:]; masked max over s -> score[h] ----
  const int slen = session_len[b];
  const int hlen = hist_len[b];
  for (int nt = 2 * w; nt < 2 * w + 2; ++nt) {
    v8f acc0 = {}, acc1 = {}, acc2 = {}, acc3 = {};
    const float* hbase = hist + (size_t)(nt * 16 + hl) * kD + (hi << 1);
    const float* qbase = sm_q + hl * XS_STRIDE + (hi << 1);
    for (int k0 = 0; k0 < kD; k0 += 4) {
      v2f bf = *(const v2f*)(hbase + k0);
      v2f a0 = *(const v2f*)(qbase + 0 * 16 * XS_STRIDE + k0);
      v2f a1 = *(const v2f*)(qbase + 1 * 16 * XS_STRIDE + k0);
      v2f a2 = *(const v2f*)(qbase + 2 * 16 * XS_STRIDE + k0);
      v2f a3 = *(const v2f*)(qbase + 3 * 16 * XS_STRIDE + k0);
      acc0 = __builtin_amdgcn_wmma_f32_16x16x4_f32(false, a0, false, bf, (short)0, acc0, false, false);
      acc1 = __builtin_amdgcn_wmma_f32_16x16x4_f32(false, a1, false, bf, (short)0, acc1, false, false);
      acc2 = __builtin_amdgcn_wmma_f32_16x16x4_f32(false, a2, false, bf, (short)0, acc2, false, false);
      acc3 = __builtin_amdgcn_wmma_f32_16x16x4_f32(false, a3, false, bf, (short)0, acc3, false, false);
    }
    // per-lane masked max over the s values this lane holds (s = m*16 + r + 8*hi)
    float mx = kNullAtt;
    #pragma unroll
    for (int r = 0; r < 8; ++r) {
      int sb = (hi << 3) + r;
      if (sb + 0  < slen) mx = fmaxf(mx, acc0[r]);
      if (sb + 16 < slen) mx = fmaxf(mx, acc1[r]);
      if (sb + 32 < slen) mx = fmaxf(mx, acc2[r]);
      if (sb + 48 < slen) mx = fmaxf(mx, acc3[r]);
    }
    // lane l and l+16 hold the same h (= nt*16 + hl) with complementary s halves
    mx = fmaxf(mx, __shfl_xor(mx, 16, 32));
    int h = nt * 16 + hl;
    if (h >= hlen) mx = kNullAtt;
    if (hi == 0) sm_score[h] = mx;
  }
  __syncthreads();

  // ---- Phase 4: softmax over h, weighted sum of hist rows ----
  float sc = sm_score[t];
  float mval = sc;
  #pragma unroll
  for (int off = 16; off > 0; off >>= 1)
    mval = fmaxf(mval, __shfl_xor(mval, off, 32));
  if (ln == 0) sm_redA[w] = mval;
  __syncthreads();
  float bmax = sm_redA[0];
  #pragma unroll
  for (int i = 1; i < 8; ++i) bmax = fmaxf(bmax, sm_redA[i]);

  float e = __expf(sc - bmax);
  float ssum = e;
  #pragma unroll
  for (int off = 16; off > 0; off >>= 1)
    ssum += __shfl_xor(ssum, off, 32);
  if (ln == 0) sm_redB[w] = ssum;
  __syncthreads();
  float tot = 0.f;
  #pragma unroll
  for (int i = 0; i < 8; ++i) tot += sm_redB[i];
  sm_w[t] = e / tot;
  __syncthreads();

  // rep[d] = sum_h w[h] * hist[h][d]; thread t = d -> coalesced hist column reads
  float acc = 0.f;
  const float* hcol = hist + t;
  #pragma unroll 4
  for (int h = 0; h < kH; ++h)
    acc += sm_w[h] * hcol[(size_t)h * kD];

  out[(size_t)b * kD + t] = acc;                        // user_hist_rep [B, D]
  out[(size_t)kB * kD + (size_t)b * kH + t] = sc;       // score         [B, H]
}

extern "C" void kernel_launch(void* const* d_in, const int* in_sizes, int n_in,
                              void* d_out, int out_size, void* d_ws, size_t ws_size,
                              hipStream_t stream) {
  (void)in_sizes; (void)n_in; (void)d_ws; (void)ws_size; (void)out_size;
  const float* item_emb    = (const float*)d_in[0];
  const float* x_session   = (const float*)d_in[1];
  const int*   session_len = (const int*)  d_in[2];
  const float* user_hist   = (const float*)d_in[3];
  const int*   hist_len    = (const int*)  d_in[4];
  const float* W1          = (const float*)d_in[5];
  const float* b1          = (const float*)d_in[6];
  float* out = (float*)d_out;

  const size_t lds_bytes = (size_t)LDS_FLOATS * sizeof(float);  // ~137 KB (<320 KB/WGP)
  hipFuncSetAttribute((const void*)coatt_wmma_kernel,
                      hipFuncAttributeMaxDynamicSharedMemorySize, (int)lds_bytes);
  coatt_wmma_kernel<<<dim3(kB), dim3(256), lds_bytes, stream>>>(
      item_emb, x_session, session_len, user_hist, hist_len, W1, b1, out);
}